// RNN_83631603188218
// MI455X (gfx1250) — compile-verified
//
#include <hip/hip_runtime.h>

// ---------------------------------------------------------------------------
// 2-layer ReLU RNN (T=256, B=64, H=1024) + FC, fp32 via V_WMMA_F32_16X16X4_F32
// ---------------------------------------------------------------------------

typedef float v2f __attribute__((ext_vector_type(2)));
typedef float v8f __attribute__((ext_vector_type(8)));

#define RNN_T 256
#define RNN_B 64
#define RNN_H 1024
#define RNN_D 256
#define REC_BLOCKS 32
#define LDS_ROWS 32
#define LDS_STRIDE 1028   // 1028 % 64 == 4 -> conflict-free b64 LDS reads
#define LDS_BYTES (LDS_ROWS * LDS_STRIDE * 4)

__device__ __forceinline__ v8f wmma4(v2f a, v2f b, v8f c) {
  // D = A(16x4,f32) * B(4x16,f32) + C(16x16,f32)
  return __builtin_amdgcn_wmma_f32_16x16x4_f32(
      /*neg_a=*/false, a, /*neg_b=*/false, b,
      /*c_mod=*/(short)0, c, /*reuse_a=*/false, /*reuse_b=*/false);
}

// ---------------------------------------------------------------------------
// Generic GEMM: C[M,N] = A[M,K] * W[N,K]^T + bias(+bias2). One wave per tile.
// amode==1: A row r maps to x[b=r&63][t=r>>6][.] (x stored [B,T,D], D=K=256)
// ---------------------------------------------------------------------------
__global__ void __launch_bounds__(256)
wmma_gemm_kernel(const float* __restrict__ A, const float* __restrict__ W,
                 const float* __restrict__ bias, const float* __restrict__ bias2,
                 float* __restrict__ C, int M, int N, int K, int amode) {
  const int wid  = (blockIdx.x * blockDim.x + threadIdx.x) >> 5;
  const int lane = threadIdx.x & 31;
  const int mt_count = M >> 4;
  const int mt = wid % mt_count;
  const int nt = wid / mt_count;
  if (nt >= (N >> 4)) return;                 // wave-uniform guard (EXEC stays all-1)

  const int l16 = lane & 15;
  const int hi2 = lane >> 4;                  // 0: K lanes 0..1, 1: K lanes 2..3
  const int mbase = mt << 4;
  const int nbase = nt << 4;

  float bsum = bias ? bias[nbase + l16] : 0.0f;
  if (bias2) bsum += bias2[nbase + l16];
  v8f c;
#pragma unroll
  for (int r = 0; r < 8; ++r) c[r] = bsum;

  const int arow = mbase + l16;
  const float* arow_ptr;
  if (amode == 1) {
    const int bb = arow & 63, tt = arow >> 6;
    arow_ptr = A + ((size_t)bb * RNN_T + tt) * RNN_D;
  } else {
    arow_ptr = A + (size_t)arow * K;
  }
  const float* wrow_ptr = W + (size_t)(nbase + l16) * K;

#pragma unroll 4
  for (int kb = 0; kb < K; kb += 4) {
    v2f av = *(const v2f*)(arow_ptr + kb + 2 * hi2);
    v2f bv = *(const v2f*)(wrow_ptr + kb + 2 * hi2);
    c = wmma4(av, bv, c);
  }

  float* crow = C + (size_t)(mbase + 8 * hi2) * N + nbase + l16;
#pragma unroll
  for (int r = 0; r < 8; ++r) crow[(size_t)r * N] = c[r];
}

// ---------------------------------------------------------------------------
// Grid-wide barrier (monotonic counter; ctr zeroed by init kernel per launch)
// ---------------------------------------------------------------------------
__device__ __forceinline__ void grid_sync(unsigned* ctr, unsigned target) {
  __threadfence();                            // release h[t] stores device-wide
  __syncthreads();
  if (threadIdx.x == 0) {
    __hip_atomic_fetch_add(ctr, 1u, __ATOMIC_ACQ_REL, __HIP_MEMORY_SCOPE_AGENT);
    while (__hip_atomic_load(ctr, __ATOMIC_ACQUIRE, __HIP_MEMORY_SCOPE_AGENT) < target) {
      __builtin_amdgcn_s_sleep(4);
    }
  }
  __syncthreads();
  __threadfence();                            // acquire for all lanes
}

__global__ void zero_ctr_kernel(unsigned* ctr) { *ctr = 0u; }

// ---------------------------------------------------------------------------
// Persistent recurrence: h[t] = relu(P[t] + h[t-1] @ Whh^T), t = 0..255.
// 32 WGs x 8 waves; WG wg owns output columns [32*wg, 32*wg+32); its Whh rows
// live in LDS for the whole kernel. Grid barrier between timesteps.
// ---------------------------------------------------------------------------
__global__ void __launch_bounds__(256, 1)
rnn_recur_kernel(const float* __restrict__ P,    // [T][B][H], biases folded in
                 const float* __restrict__ Whh,  // [H][H]
                 float* __restrict__ Hbuf,       // [T][B][H] out (and h state)
                 unsigned* __restrict__ ctr) {
  extern __shared__ float lds_w[];               // [32][LDS_STRIDE]
  const int wg  = blockIdx.x;                    // 0..31
  const int tid = threadIdx.x;

  // Stage Whh rows [32*wg, 32*wg+32) into LDS (coalesced, once per kernel).
  {
    const float* src = Whh + (size_t)wg * LDS_ROWS * RNN_H;
    for (int i = tid; i < LDS_ROWS * RNN_H; i += 256) {
      const int nl = i >> 10, k = i & (RNN_H - 1);
      lds_w[nl * LDS_STRIDE + k] = src[i];
    }
  }
  __syncthreads();

  const int w    = tid >> 5;                     // wave 0..7
  const int lane = tid & 31;
  const int l16  = lane & 15;
  const int hi2  = lane >> 4;
  const int mt   = w & 3;                        // batch tile 0..3 (B=64)
  const int ntl  = w >> 2;                       // 0..1 (local 16-col group)
  const int mbase = mt << 4;
  const int nbase = wg * 32 + (ntl << 4);

  const float* lds_b = lds_w + (size_t)((ntl << 4) + l16) * LDS_STRIDE + 2 * hi2;

  for (int t = 0; t < RNN_T; ++t) {
    // C tile <- precomputed input projection (includes b_ih + b_hh)
    const float* prow =
        P + ((size_t)t * RNN_B + mbase + 8 * hi2) * RNN_H + nbase + l16;
    v8f c;
#pragma unroll
    for (int r = 0; r < 8; ++r) c[r] = prow[(size_t)r * RNN_H];

    if (t > 0) {                                 // h[-1] == 0: skip matmul at t=0
      const float* arow =
          Hbuf + ((size_t)(t - 1) * RNN_B + mbase + l16) * RNN_H + 2 * hi2;
#pragma unroll 4
      for (int kb = 0; kb < RNN_H; kb += 4) {
        v2f av = *(const v2f*)(arow + kb);       // global_load_b64 from L2
        v2f bv = *(const v2f*)(lds_b + kb);      // ds_load_b64, bank-conflict-free
        c = wmma4(av, bv, c);
      }
    }

    float* hrow =
        Hbuf + ((size_t)t * RNN_B + mbase + 8 * hi2) * RNN_H + nbase + l16;
#pragma unroll
    for (int r = 0; r < 8; ++r)
      hrow[(size_t)r * RNN_H] = c[r] > 0.0f ? c[r] : 0.0f;

    grid_sync(ctr, (unsigned)REC_BLOCKS * (unsigned)(t + 1));
  }
}

// ---------------------------------------------------------------------------
extern "C" void kernel_launch(void* const* d_in, const int* in_sizes, int n_in,
                              void* d_out, int out_size, void* d_ws, size_t ws_size,
                              hipStream_t stream) {
  const float* x     = (const float*)d_in[0];
  const float* W_ih0 = (const float*)d_in[1];
  const float* W_hh0 = (const float*)d_in[2];
  const float* b_ih0 = (const float*)d_in[3];
  const float* b_hh0 = (const float*)d_in[4];
  const float* W_ih1 = (const float*)d_in[5];
  const float* W_hh1 = (const float*)d_in[6];
  const float* b_ih1 = (const float*)d_in[7];
  const float* b_hh1 = (const float*)d_in[8];
  const float* W_fc  = (const float*)d_in[9];
  const float* b_fc  = (const float*)d_in[10];
  float* out = (float*)d_out;

  char* ws = (char*)d_ws;
  float*    Pbuf = (float*)ws;                          // 64 MiB (P0, then P1)
  float*    Hbuf = (float*)(ws + ((size_t)64 << 20));   // 64 MiB (h0, then h1)
  unsigned* ctr  = (unsigned*)(ws + ((size_t)128 << 20));

  // Allow >64KB dynamic LDS for the persistent kernel (idempotent; rc ignored).
  (void)hipFuncSetAttribute((const void*)rnn_recur_kernel,
                            hipFuncAttributeMaxDynamicSharedMemorySize,
                            (int)LDS_BYTES);

  const int TB = RNN_T * RNN_B;                          // 16384 rows

  // 1) P0 = x @ W_ih0^T + b_ih0 + b_hh0   (M=16384, K=256, N=1024)
  {
    const int tiles = (TB >> 4) * (RNN_H >> 4);          // 65536
    wmma_gemm_kernel<<<tiles / 8, 256, 0, stream>>>(
        x, W_ih0, b_ih0, b_hh0, Pbuf, TB, RNN_H, RNN_D, /*amode=*/1);
  }

  // 2) layer-0 recurrence -> Hbuf = h0[T][B][H]
  zero_ctr_kernel<<<1, 1, 0, stream>>>(ctr);
  rnn_recur_kernel<<<REC_BLOCKS, 256, LDS_BYTES, stream>>>(Pbuf, W_hh0, Hbuf, ctr);

  // 3) P1 = h0 @ W_ih1^T + b_ih1 + b_hh1  (M=16384, K=1024, N=1024)
  {
    const int tiles = (TB >> 4) * (RNN_H >> 4);
    wmma_gemm_kernel<<<tiles / 8, 256, 0, stream>>>(
        Hbuf, W_ih1, b_ih1, b_hh1, Pbuf, TB, RNN_H, RNN_H, /*amode=*/0);
  }

  // 4) layer-1 recurrence -> Hbuf = h1 (h0 is dead, safe to overwrite)
  zero_ctr_kernel<<<1, 1, 0, stream>>>(ctr);
  rnn_recur_kernel<<<REC_BLOCKS, 256, LDS_BYTES, stream>>>(Pbuf, W_hh1, Hbuf, ctr);

  // 5) out[B,1,256] = h1[T-1] @ W_fc^T + b_fc  (M=64, K=1024, N=256)
  {
    const float* h_last = Hbuf + (size_t)(RNN_T - 1) * RNN_B * RNN_H;
    const int tiles = (RNN_B >> 4) * (RNN_D >> 4);       // 64 tiles -> 8 blocks
    wmma_gemm_kernel<<<tiles / 8, 256, 0, stream>>>(
        h_last, W_fc, b_fc, nullptr, out, RNN_B, RNN_D, RNN_H, /*amode=*/0);
  }
}